// SelfAttention_16870631539114
// MI455X (gfx1250) — compile-verified
//
#include <hip/hip_runtime.h>
#include <hip/hip_bf16.h>
#include <math.h>

typedef float v2f __attribute__((ext_vector_type(2)));
typedef float v8f __attribute__((ext_vector_type(8)));
typedef int   v4i_vs __attribute__((vector_size(4 * sizeof(int))));

#define DIM 4096
#define NH 32
#define NKV 8
#define HD 128
#define NREP 4
#define BS 16
#define SS 8
#define MROWS 128   /* BS*SS */
#define MAXSEQ 4096
#define NSEG 4      /* flash-decoding segments over T */

__device__ __forceinline__ v8f wmma_f32(v2f a, v2f b, v8f c) {
  // D = A(16x4 f32) x B(4x16 f32) + C(16x16 f32)
  return __builtin_amdgcn_wmma_f32_16x16x4_f32(
      false, a, false, b, (short)0, c, false, false);
}

// ---- CDNA5 async global->LDS copy (16B per lane), ASYNCcnt tracked --------
#if __has_builtin(__builtin_amdgcn_global_load_async_to_lds_b128)
#define HAS_ASYNC_LDS 1
#else
#define HAS_ASYNC_LDS 0
#endif

typedef __attribute__((address_space(1))) void* as1_void;
typedef __attribute__((address_space(3))) void* as3_void;
typedef __attribute__((address_space(1))) v4i_vs* as1_v4;
typedef __attribute__((address_space(3))) v4i_vs* as3_v4;

__device__ __forceinline__ void async_cp16(const float* g, float* l) {
#if HAS_ASYNC_LDS
  as1_void gv = (as1_void)const_cast<float*>(g);
  as3_void lv = (as3_void)l;
  __builtin_amdgcn_global_load_async_to_lds_b128((as1_v4)gv, (as3_v4)lv, 0, 0);
#else
  *(float4*)l = *(const float4*)g;
#endif
}

__device__ __forceinline__ void async_wait_all() {
#if HAS_ASYNC_LDS
#if __has_builtin(__builtin_amdgcn_s_wait_asynccnt)
  __builtin_amdgcn_s_wait_asynccnt(0);
#else
  asm volatile("s_wait_asynccnt 0" ::: "memory");
#endif
#endif
}

// ---------------------------------------------------------------------------
// NT GEMM:  C[M,N] = A[M,K] * W[N,K]^T  (fp32 WMMA, async-LDS double buffer)
// block = 128 threads (4 waves), tile = 16(M) x 64(N), K chunk = 32.
// ---------------------------------------------------------------------------
#define GKC 32
#define GSTR 36   // LDS row stride (floats): 16B-aligned rows, conflict-free

__global__ __launch_bounds__(128)
void gemm_nt_wmma(const float* __restrict__ A, const float* __restrict__ W,
                  float* __restrict__ C, int K, int ldc)
{
  __shared__ float As[2][16 * GSTR];
  __shared__ float Bs[2][64 * GSTR];
  const int tid  = threadIdx.x;
  const int lane = tid & 31;
  const int w    = tid >> 5;
  const int lm   = lane & 15;
  const int off  = (lane >> 4) * 2;
  const int mh   = 8 * (lane >> 4);
  const int m0 = blockIdx.y * 16;
  const int n0 = blockIdx.x * 64;
  const int ra = tid >> 3;           // A tile: 8 float4 per 32-wide row
  const int ca = (tid & 7) * 4;

  v8f acc;
#pragma unroll
  for (int j = 0; j < 8; ++j) acc[j] = 0.0f;

  // stage chunk 0 into buffer 0
  async_cp16(A + (size_t)(m0 + ra) * K + ca, &As[0][ra * GSTR + ca]);
#pragma unroll
  for (int i = 0; i < 4; ++i) {
    int idx = tid + i * 128;
    int r = idx >> 3, c = (idx & 7) * 4;
    async_cp16(W + (size_t)(n0 + r) * K + c, &Bs[0][r * GSTR + c]);
  }
  async_wait_all();
  __syncthreads();

  int cur = 0;
  for (int k0 = 0; k0 < K; k0 += GKC) {
    const int nxt = k0 + GKC;
    if (nxt < K) {  // prefetch next chunk into the other buffer (async)
      async_cp16(A + (size_t)(m0 + ra) * K + nxt + ca,
                 &As[cur ^ 1][ra * GSTR + ca]);
#pragma unroll
      for (int i = 0; i < 4; ++i) {
        int idx = tid + i * 128;
        int r = idx >> 3, c = (idx & 7) * 4;
        async_cp16(W + (size_t)(n0 + r) * K + nxt + c,
                   &Bs[cur ^ 1][r * GSTR + c]);
      }
    }
    const float* ap = &As[cur][lm * GSTR + off];
    const float* bp = &Bs[cur][(w * 16 + lm) * GSTR + off];
#pragma unroll
    for (int kk = 0; kk < GKC; kk += 4) {
      v2f a = *(const v2f*)(ap + kk);
      v2f b = *(const v2f*)(bp + kk);
      acc = wmma_f32(a, b, acc);
    }
    async_wait_all();
    __syncthreads();
    cur ^= 1;
  }
  const int n = n0 + w * 16 + lm;
#pragma unroll
  for (int j = 0; j < 8; ++j)
    C[(size_t)(m0 + j + mh) * ldc + n] = acc[j];
}

// ---------------------------------------------------------------------------
// RoPE (in-place on workspace q / k). X layout: [MROWS][n_heads*HD]
// ---------------------------------------------------------------------------
__global__ void rope_kernel(float* __restrict__ X, int n_heads,
                            const int* __restrict__ sp_ptr)
{
  int idx = blockIdx.x * blockDim.x + threadIdx.x;
  int total = MROWS * n_heads * (HD / 2);
  if (idx >= total) return;
  int d2 = idx & (HD / 2 - 1);
  int h  = (idx >> 6) % n_heads;
  int m  = idx / ((HD / 2) * n_heads);
  int s  = m & (SS - 1);
  float pos  = (float)(*sp_ptr + s);
  float freq = pos * powf(10000.0f, -(float)(2 * d2) / (float)HD);
  float sn, cs;
  sincosf(freq, &sn, &cs);
  float* p = X + ((size_t)m * n_heads + h) * HD + 2 * d2;
  float x0 = p[0], x1 = p[1];
  p[0] = x0 * cs - x1 * sn;
  p[1] = x0 * sn + x1 * cs;
}

// ---------------------------------------------------------------------------
// Flash-decoding attention, stage 1: per (kv-head g, batch b, segment z).
// 32 query rows (m = s*4 + r) x one T-segment.  Online softmax, fp32 WMMA.
// Emits unnormalized acc (32x128) + per-row running max m / sum l.
// ---------------------------------------------------------------------------
#define TC 32
#define KSTR 132  // LDS row stride for Q/K/V tiles
#define PSTR 36   // LDS row stride for the 32xTC probability tile

__global__ __launch_bounds__(128)
void attn_wmma(const float* __restrict__ q_ws, const float* __restrict__ k_ws,
               const float* __restrict__ v_ws, const float* __restrict__ cache_k,
               const float* __restrict__ cache_v,
               float* __restrict__ pacc, float* __restrict__ pm,
               float* __restrict__ pl, const int* __restrict__ sp_ptr)
{
  __shared__ float Qs[32 * KSTR];
  __shared__ float Ks[TC * KSTR];
  __shared__ float Vs[TC * KSTR];
  __shared__ float Ps[32 * PSTR];
  __shared__ float mrow[32], lrow[32], arow[32];

  const int tid = threadIdx.x, lane = tid & 31, w = tid >> 5;
  const int lm = lane & 15, off = (lane >> 4) * 2, mh = 8 * (lane >> 4);
  const int g = blockIdx.x, b = blockIdx.y, z = blockIdx.z;
  const int sp = *sp_ptr;
  const int T  = sp + SS;
  const int mi = w >> 1;   // query-row tile (0..1)
  const int nj = w & 1;    // key sub-tile within chunk (0..1)

  // segment chunk range
  const int nchunk = (T + TC - 1) / TC;
  const int per    = (nchunk + NSEG - 1) / NSEG;
  const int c0 = z * per;
  const int c1 = (c0 + per < nchunk) ? (c0 + per) : nchunk;

  // --- stage Q (32 x 128), row m = s*4 + r (async to LDS) ---
  for (int i = tid; i < 32 * 32; i += 128) {
    int row = i >> 5;
    int c = (i & 31) * 4;
    int s = row >> 2, r = row & 3;
    async_cp16(q_ws + (size_t)(b * SS + s) * DIM + (g * NREP + r) * HD + c,
               Qs + row * KSTR + c);
  }
  if (tid < 32) { mrow[tid] = -INFINITY; lrow[tid] = 0.0f; }

  v8f acc[4];
#pragma unroll
  for (int i = 0; i < 4; ++i)
#pragma unroll
    for (int j = 0; j < 8; ++j) acc[i][j] = 0.0f;

  async_wait_all();
  __syncthreads();

  for (int ci = c0; ci < c1; ++ci) {
    const int tc = ci * TC;
    // ---- stage K/V chunk (TC x 128) async: cache for t<sp, fresh otherwise ----
    for (int i = tid; i < TC * 32; i += 128) {
      int row = i >> 5;
      int c = (i & 31) * 4;
      int t = tc + row;
      int tcl = (t < MAXSEQ) ? t : (MAXSEQ - 1);
      bool fresh = (t >= sp) && (t < T);
      int tw = fresh ? (t - sp) : 0;
      size_t cbase = (((size_t)b * MAXSEQ + tcl) * NKV + g) * HD + c;
      size_t wbase = ((size_t)(b * SS + tw) * NKV + g) * HD + c;
      const float* sk = fresh ? (k_ws + wbase) : (cache_k + cbase);
      const float* sv = fresh ? (v_ws + wbase) : (cache_v + cbase);
      async_cp16(sk, Ks + row * KSTR + c);
      async_cp16(sv, Vs + row * KSTR + c);
    }
    async_wait_all();
    __syncthreads();

    // ---- scores: S(mi,nj) = Q(mi) . K(nj)^T, scaled ----
    v8f sc;
#pragma unroll
    for (int j = 0; j < 8; ++j) sc[j] = 0.0f;
    {
      const float* ap = Qs + (mi * 16 + lm) * KSTR + off;
      const float* bp = Ks + (nj * 16 + lm) * KSTR + off;
      for (int kk = 0; kk < HD; kk += 4) {
        v2f a  = *(const v2f*)(ap + kk);
        v2f bb = *(const v2f*)(bp + kk);
        sc = wmma_f32(a, bb, sc);
      }
    }
    const float scale = 0.08838834764831845f;  // 1/sqrt(128)
#pragma unroll
    for (int j = 0; j < 8; ++j)
      Ps[(mi * 16 + j + mh) * PSTR + nj * 16 + lm] = sc[j] * scale;
    __syncthreads();

    // ---- online softmax: one thread per query row ----
    if (tid < 32) {
      int row = tid;
      float mx = -INFINITY;
      for (int c = 0; c < TC; ++c)
        if (tc + c < T) mx = fmaxf(mx, Ps[row * PSTR + c]);
      float mold  = mrow[row];
      float mnew  = fmaxf(mold, mx);
      float alpha = expf(mold - mnew);
      float sum = 0.0f;
      for (int c = 0; c < TC; ++c) {
        float p = (tc + c < T) ? expf(Ps[row * PSTR + c] - mnew) : 0.0f;
        Ps[row * PSTR + c] = p;
        sum += p;
      }
      lrow[row] = lrow[row] * alpha + sum;
      mrow[row] = mnew;
      arow[row] = alpha;
    }
    __syncthreads();

    // ---- rescale accumulators, then acc += P(32xTC) . V(TCx128) ----
    float al[8];
#pragma unroll
    for (int j = 0; j < 8; ++j) al[j] = arow[mi * 16 + j + mh];
#pragma unroll
    for (int i = 0; i < 4; ++i)
#pragma unroll
      for (int j = 0; j < 8; ++j) acc[i][j] *= al[j];

    {
      const float* ap = Ps + (mi * 16 + lm) * PSTR + off;
      for (int kk = 0; kk < TC; kk += 4) {
        v2f a = *(const v2f*)(ap + kk);
        int t0 = kk + off;
#pragma unroll
        for (int i = 0; i < 4; ++i) {
          int n = (4 * (w & 1) + i) * 16 + lm;
          v2f bb;
          bb.x = Vs[t0 * KSTR + n];
          bb.y = Vs[(t0 + 1) * KSTR + n];
          acc[i] = wmma_f32(a, bb, acc[i]);
        }
      }
    }
    __syncthreads();
  }

  // ---- emit segment partials (unnormalized) ----
  const size_t pb = ((size_t)(z * BS + b) * NKV + g) * 32;
  if (tid < 32) { pm[pb + tid] = mrow[tid]; pl[pb + tid] = lrow[tid]; }
#pragma unroll
  for (int i = 0; i < 4; ++i) {
    int n = (4 * (w & 1) + i) * 16 + lm;
#pragma unroll
    for (int j = 0; j < 8; ++j) {
      int m = mi * 16 + j + mh;
      pacc[(pb + m) * HD + n] = acc[i][j];
    }
  }
}

// ---------------------------------------------------------------------------
// Flash-decoding stage 2: merge NSEG segment partials -> ctx[b][s][h][d]
// ---------------------------------------------------------------------------
__global__ void attn_combine(const float* __restrict__ pacc,
                             const float* __restrict__ pm,
                             const float* __restrict__ pl,
                             float* __restrict__ ctx)
{
  int idx = blockIdx.x * blockDim.x + threadIdx.x;
  const int total = BS * NKV * 32 * HD;
  if (idx >= total) return;
  int d = idx & (HD - 1);
  int m = (idx >> 7) & 31;
  int g = (idx >> 12) & (NKV - 1);
  int b = idx >> 15;

  float M = -INFINITY;
#pragma unroll
  for (int z = 0; z < NSEG; ++z) {
    size_t pb = ((size_t)(z * BS + b) * NKV + g) * 32;
    M = fmaxf(M, pm[pb + m]);
  }
  float L = 0.0f, num = 0.0f;
#pragma unroll
  for (int z = 0; z < NSEG; ++z) {
    size_t pb = ((size_t)(z * BS + b) * NKV + g) * 32;
    float wgt = expf(pm[pb + m] - M);
    L   += wgt * pl[pb + m];
    num += wgt * pacc[(pb + m) * HD + d];
  }
  int s = m >> 2, r = m & 3;
  ctx[(size_t)(b * SS + s) * DIM + (g * NREP + r) * HD + d] = num / L;
}

// ---------------------------------------------------------------------------
extern "C" void kernel_launch(void* const* d_in, const int* in_sizes, int n_in,
                              void* d_out, int out_size, void* d_ws, size_t ws_size,
                              hipStream_t stream)
{
  const float* x  = (const float*)d_in[0];
  const float* Wq = (const float*)d_in[1];
  const float* Wk = (const float*)d_in[2];
  const float* Wv = (const float*)d_in[3];
  const float* Wo = (const float*)d_in[4];
  const float* ck = (const float*)d_in[5];
  const float* cv = (const float*)d_in[6];
  const int*   sp = (const int*)d_in[7];
  float* out = (float*)d_out;

  float* ws   = (float*)d_ws;
  float* q_ws = ws;                                     // 128*4096
  float* k_ws = q_ws + (size_t)MROWS * DIM;             // 128*1024
  float* v_ws = k_ws + (size_t)MROWS * NKV * HD;        // 128*1024
  float* c_ws = v_ws + (size_t)MROWS * NKV * HD;        // 128*4096
  float* pacc = c_ws + (size_t)MROWS * DIM;             // NSEG*16*8*32*128
  float* pm   = pacc + (size_t)NSEG * BS * NKV * 32 * HD;
  float* pl   = pm + (size_t)NSEG * BS * NKV * 32;

  dim3 blk(128);
  // QKV projections
  gemm_nt_wmma<<<dim3(DIM / 64, MROWS / 16), blk, 0, stream>>>(x, Wq, q_ws, DIM, DIM);
  gemm_nt_wmma<<<dim3((NKV * HD) / 64, MROWS / 16), blk, 0, stream>>>(x, Wk, k_ws, DIM, NKV * HD);
  gemm_nt_wmma<<<dim3((NKV * HD) / 64, MROWS / 16), blk, 0, stream>>>(x, Wv, v_ws, DIM, NKV * HD);
  // RoPE on q and k
  rope_kernel<<<(MROWS * NH * (HD / 2) + 255) / 256, 256, 0, stream>>>(q_ws, NH, sp);
  rope_kernel<<<(MROWS * NKV * (HD / 2) + 255) / 256, 256, 0, stream>>>(k_ws, NKV, sp);
  // flash-decoding attention: (kv-head, batch, segment) grid, then combine
  attn_wmma<<<dim3(NKV, BS, NSEG), blk, 0, stream>>>(q_ws, k_ws, v_ws, ck, cv,
                                                     pacc, pm, pl, sp);
  attn_combine<<<(BS * NKV * 32 * HD + 255) / 256, 256, 0, stream>>>(pacc, pm, pl, c_ws);
  // output projection
  gemm_nt_wmma<<<dim3(DIM / 64, MROWS / 16), blk, 0, stream>>>(c_ws, Wo, out, DIM, DIM);
}